// TSSA_3461743640708
// MI455X (gfx1250) — compile-verified
//
#include <hip/hip_runtime.h>
#include <math.h>

#define N_ROWS 8192
#define M_COLS 4096
#define ROWS_PER_CHUNK 64
#define NUM_CHUNKS 128          // 8192 / 64
#define COL4 1024               // M_COLS / 4

typedef __attribute__((ext_vector_type(2))) float v2f;
typedef __attribute__((ext_vector_type(8))) float v8f;

// ---------------- k0: column-0 diffs ----------------
__global__ void k0_da(const float* __restrict__ x, float* __restrict__ da) {
    int i = blockIdx.x * blockDim.x + threadIdx.x;
    if (i < N_ROWS - 1)
        da[i] = x[(size_t)(i + 1) * M_COLS] - x[(size_t)i * M_COLS];
}

__device__ __forceinline__ float hpdd_f(float dai, float c, float n) {
    float db = n - c;
    float df = dai - db;
    return (dai * db >= 0.0f) ? df : fabsf(df);
}

// ---------------- k1: streaming partial reductions ----------------
// grid = (4 column-chunks, 128 row-chunks), block = 256 threads, float4 per thread.
__global__ void k1_partial(const float* __restrict__ x, const float* __restrict__ da,
                           float* __restrict__ ptss, float* __restrict__ pcs) {
    const float4* x4 = (const float4*)x;
    int col4 = blockIdx.x * blockDim.x + threadIdx.x;   // 0..1023 (float4 units)
    int r0 = blockIdx.y * ROWS_PER_CHUNK;

    float4 cur = x4[(size_t)r0 * COL4 + col4];
    float tsx = 0.f, tsy = 0.f, tsz = 0.f, tsw = 0.f;
    float csx = 0.f, csy = 0.f, csz = 0.f, csw = 0.f;

    int steps = (N_ROWS - 1) - r0;
    if (steps > ROWS_PER_CHUNK) steps = ROWS_PER_CHUNK;

    for (int s = 0; s < steps; ++s) {
        int i = r0 + s;
        float4 nxt = x4[(size_t)(i + 1) * COL4 + col4];
        int ip = i + 2; if (ip > N_ROWS - 1) ip = N_ROWS - 1;
        __builtin_prefetch(&x4[(size_t)ip * COL4 + col4], 0, 0);
        float dai = da[i];                 // block-uniform
        csx += cur.x; csy += cur.y; csz += cur.z; csw += cur.w;
        tsx += hpdd_f(dai, cur.x, nxt.x);
        tsy += hpdd_f(dai, cur.y, nxt.y);
        tsz += hpdd_f(dai, cur.z, nxt.z);
        tsw += hpdd_f(dai, cur.w, nxt.w);
        cur = nxt;
    }
    // last row-chunk: row 8191 contributes to colsum only (at most one iteration)
    for (int s = steps; s < ROWS_PER_CHUNK; ++s) {
        csx += cur.x; csy += cur.y; csz += cur.z; csw += cur.w;
    }

    size_t base = (size_t)blockIdx.y * M_COLS;
    float4 t4; t4.x = tsx; t4.y = tsy; t4.z = tsz; t4.w = tsw;
    float4 c4; c4.x = csx; c4.y = csy; c4.z = csz; c4.w = csw;
    ((float4*)(ptss + base))[col4] = t4;
    ((float4*)(pcs  + base))[col4] = c4;
}

// ---------------- k2: fp32 WMMA tree reduction over the 128 row-chunks ----------------
// D = A(16x4) * ones(4x16) + C  => D[m,n] = sum_k A[m,k] + C[m,n]  (exact fp32)
// Each wave handles 16 columns; block = 8 waves = 128 columns; grid = 32 blocks.
__global__ void k2_reduce_wmma(const float* __restrict__ ptss, const float* __restrict__ pcs,
                               float* __restrict__ tss, float* __restrict__ cs) {
    int lane = threadIdx.x & 31;
    int wave = threadIdx.x >> 5;
    int col0 = blockIdx.x * 128 + wave * 16;
    int m  = lane & 15;
    int kb = (lane >> 4) << 1;              // lanes 0-15 -> K=0,1 ; lanes 16-31 -> K=2,3
    int col = col0 + m;

    v2f bone; bone.x = 1.0f; bone.y = 1.0f; // B = ones (any K x N layout works)
    v8f ct = {};
    v8f cc = {};

    for (int cb = 0; cb < NUM_CHUNKS; cb += 4) {
        v2f at, ac;
        at.x = ptss[(size_t)(cb + kb + 0) * M_COLS + col];
        at.y = ptss[(size_t)(cb + kb + 1) * M_COLS + col];
        ac.x = pcs [(size_t)(cb + kb + 0) * M_COLS + col];
        ac.y = pcs [(size_t)(cb + kb + 1) * M_COLS + col];
        ct = __builtin_amdgcn_wmma_f32_16x16x4_f32(false, at, false, bone, (short)0, ct, false, false);
        cc = __builtin_amdgcn_wmma_f32_16x16x4_f32(false, ac, false, bone, (short)0, cc, false, false);
    }

    // C/D layout: lane 0 holds M=0..7 (N=0) in v[0..7]; lane 16 holds M=8..15 (N=0).
    if (lane == 0) {
        for (int r = 0; r < 8; ++r) { tss[col0 + r] = ct[r]; cs[col0 + r] = cc[r]; }
    } else if (lane == 16) {
        for (int r = 0; r < 8; ++r) { tss[col0 + 8 + r] = ct[r]; cs[col0 + 8 + r] = cc[r]; }
    }
}

// ---------------- k3: scalar chain + final reduction ----------------
__global__ void k3_final(const float* __restrict__ tss, const float* __restrict__ cs,
                         const float* __restrict__ var, float* __restrict__ out) {
    __shared__ float s1[1024];
    __shared__ float s2[1024];
    int tid = threadIdx.x;
    float v = var[0];
    float coef   = 1.0f / (sqrtf(2.0f * 3.14159265358979323846f) * v);
    float inv2v2 = 1.0f / (2.0f * v * v);

    float st = 0.0f, sty = 0.0f;
    for (int c = tid; c < M_COLS; c += 1024) {
        if (c >= 1) {                       // tss index j corresponds to column j; skip col 0
            float ts = tss[c];
            float sc = coef * expf(-ts * ts * inv2v2);
            float t  = tanf(sc);
            st  += t;
            sty += t * cs[c];
        }
    }
    s1[tid] = st; s2[tid] = sty;
    __syncthreads();
    for (int off = 512; off > 0; off >>= 1) {
        if (tid < off) { s1[tid] += s1[tid + off]; s2[tid] += s2[tid + off]; }
        __syncthreads();
    }
    if (tid == 0) out[0] = s2[0] / s1[0];
}

extern "C" void kernel_launch(void* const* d_in, const int* in_sizes, int n_in,
                              void* d_out, int out_size, void* d_ws, size_t ws_size,
                              hipStream_t stream) {
    const float* x   = (const float*)d_in[0];
    const float* var = (const float*)d_in[1];

    // workspace layout (floats): da[8192] | ptss[128*4096] | pcs[128*4096] | tss[4096] | cs[4096]
    float* w    = (float*)d_ws;
    float* da   = w;
    float* ptss = w + 8192;
    float* pcs  = ptss + (size_t)NUM_CHUNKS * M_COLS;
    float* tss  = pcs  + (size_t)NUM_CHUNKS * M_COLS;
    float* cs   = tss  + M_COLS;

    k0_da<<<32, 256, 0, stream>>>(x, da);
    dim3 g1(4, NUM_CHUNKS);
    k1_partial<<<g1, 256, 0, stream>>>(x, da, ptss, pcs);
    k2_reduce_wmma<<<32, 256, 0, stream>>>(ptss, pcs, tss, cs);
    k3_final<<<1, 1024, 0, stream>>>(tss, cs, var, (float*)d_out);
}